// OPTCacheFlowAttention_32882269618237
// MI455X (gfx1250) — compile-verified
//
#include <hip/hip_runtime.h>

typedef __attribute__((ext_vector_type(2))) float v2f;
typedef __attribute__((ext_vector_type(4))) float v4f;
typedef __attribute__((ext_vector_type(8))) float v8f;

#define B_      32
#define H_      16
#define D_      128
#define BS_     16
#define MAXB_   64
#define S_      1024
#define SCALE_  0.08838834764831845f
#define NWAVES  8
#define NTHREADS 256

__global__ __launch_bounds__(NTHREADS) void pa_decode_kernel(
    const float* __restrict__ q,       // [B,H,D]
    const float* __restrict__ knew,    // [B,H,D]
    const float* __restrict__ vnew,    // [B,H,D]
    const float* __restrict__ kcache,  // [NB,H,D/8,BS,8]
    const float* __restrict__ vcache,  // [NB,H,BS,D]
    const int*   __restrict__ btab,    // [B,MAXB]
    const int*   __restrict__ ctxlen,  // [B]
    float*       __restrict__ out)     // [B,H*D]
{
    __shared__ float sq[D_];
    __shared__ float ssc[S_];
    __shared__ float sred[NWAVES];
    __shared__ float sbcast[2];
    __shared__ float sout[D_];

    const int bh   = blockIdx.x;
    const int b    = bh / H_;
    const int h    = bh % H_;
    const int tid  = threadIdx.x;
    const int wave = tid >> 5;
    const int lane = tid & 31;
    const int ctx  = ctxlen[b];
    const int last = ctx - 1;

    // ---- load q (pre-scaled) and zero the output accumulator ----
    if (tid < D_) {
        sq[tid]   = q[(b * H_ + h) * D_ + tid] * SCALE_;
        sout[tid] = 0.0f;
    }
    __syncthreads();

    const int hi = lane >> 4;   // lane group: K-halves of WMMA A/B frags
    const int m  = lane & 15;   // token row within tile

    // ---- hoist B fragments: q broadcast across all 16 columns ----
    v2f qf[32];
#pragma unroll
    for (int j = 0; j < 32; ++j) {
        const int d = 4 * j + 2 * hi;
        qf[j] = *(const v2f*)(sq + d);   // lanes 0-15: K={4j,4j+1}, lanes 16-31: K={4j+2,4j+3}
    }

    const int nchunks = (ctx + BS_ - 1) / BS_;   // one chunk == one cache block of 16 tokens

    // ================= Phase 1: scores = (q*scale) . K  via WMMA f32 16x16x4 =================
    for (int c = wave; c < nchunks; c += NWAVES) {
        const int blk = btab[b * MAXB_ + c];
        const float* kb = kcache + ((size_t)blk * H_ + h) * (size_t)(16 * 16 * 8);

        // prefetch next block this wave will consume
        const int cn = c + NWAVES;
        if (cn < nchunks) {
            const int blkn = btab[b * MAXB_ + cn];
            const float* kbn = kcache + ((size_t)blkn * H_ + h) * (size_t)(16 * 16 * 8);
            __builtin_prefetch(kbn + lane * 64, 0, 1);
        }

        v8f acc = {};
#pragma unroll
        for (int j = 0; j < 32; ++j) {
            // A-frag: token row m, K elements d0,d0+1 ; key layout [d/8][t][d%8]
            const int d0 = 4 * j + 2 * hi;
            const float* ap = kb + ((d0 >> 3) * 128 + m * 8 + (d0 & 7));
            v2f af = *(const v2f*)ap;    // global_load_b64, even-float aligned
            acc = __builtin_amdgcn_wmma_f32_16x16x4_f32(
                false, af, false, qf[j], (short)0, acc, false, false);
        }

        // All 16 D-columns identical (B broadcast); column N=0 lives in lanes 0 and 16.
        if (m == 0) {
            const int pbase = c * BS_ + hi * 8;
#pragma unroll
            for (int r = 0; r < 8; ++r) {
                const int p = pbase + r;
                float sc = acc[r];
                if (p >= ctx) sc = -1e9f;
                ssc[p] = sc;
            }
        }
    }
    __syncthreads();

    // ---- override score[last]: new token's K is NOT in the cache (no input mutation) ----
    if (wave == 0) {
        float part = 0.0f;
#pragma unroll
        for (int k = 0; k < 4; ++k) {
            const int d = lane * 4 + k;
            part += sq[d] * knew[(b * H_ + h) * D_ + d];
        }
        for (int off = 16; off > 0; off >>= 1) part += __shfl_xor(part, off, 32);
        if (lane == 0) ssc[last] = part;
    }
    __syncthreads();

    // ================= Phase 2: softmax over ssc[0..ctx) =================
    float lmax = -1e30f;
    for (int p = tid; p < ctx; p += NTHREADS) lmax = fmaxf(lmax, ssc[p]);
    for (int off = 16; off > 0; off >>= 1) lmax = fmaxf(lmax, __shfl_xor(lmax, off, 32));
    if (lane == 0) sred[wave] = lmax;
    __syncthreads();
    if (tid == 0) {
        float mx = sred[0];
        for (int w = 1; w < NWAVES; ++w) mx = fmaxf(mx, sred[w]);
        sbcast[0] = mx;
    }
    __syncthreads();
    const float gmax = sbcast[0];

    float lsum = 0.0f;
    for (int p = tid; p < ctx; p += NTHREADS) {
        const float e = __expf(ssc[p] - gmax);
        ssc[p] = e;
        lsum += e;
    }
    for (int off = 16; off > 0; off >>= 1) lsum += __shfl_xor(lsum, off, 32);
    if (lane == 0) sred[wave] = lsum;
    __syncthreads();
    if (tid == 0) {
        float s = 0.0f;
        for (int w = 0; w < NWAVES; ++w) s += sred[w];
        sbcast[1] = 1.0f / s;
    }
    __syncthreads();
    const float inv = sbcast[1];

    // ================= Phase 3: out = sum_p attn[p] * V[p,:] =================
    const int dl = lane * 4;                 // each lane owns 4 contiguous d's
    v4f acc4 = {0.0f, 0.0f, 0.0f, 0.0f};
    const float* vlast = vnew + (b * H_ + h) * D_;

    for (int c = wave; c < nchunks; c += NWAVES) {
        const int blk = btab[b * MAXB_ + c];
        const float* vb = vcache + ((size_t)blk * H_ + h) * (size_t)(BS_ * D_);
        const int pbase = c * BS_;
        const int rem = ctx - pbase;
        const int tmax = rem < BS_ ? rem : BS_;
        for (int t = 0; t < tmax; ++t) {
            const int p = pbase + t;
            const float coef = ssc[p];                         // LDS broadcast read
            const float* vr = (p == last) ? vlast : (vb + t * D_);
            v4f vv = *(const v4f*)(vr + dl);                   // global_load_b128, coalesced
            acc4 += vv * coef;
        }
    }

    // cross-wave combine via LDS float atomics (ds_add_f32)
    atomicAdd(&sout[dl + 0], acc4.x);
    atomicAdd(&sout[dl + 1], acc4.y);
    atomicAdd(&sout[dl + 2], acc4.z);
    atomicAdd(&sout[dl + 3], acc4.w);
    __syncthreads();

    if (tid < D_) out[(b * H_ + h) * D_ + tid] = sout[tid] * inv;
}

extern "C" void kernel_launch(void* const* d_in, const int* in_sizes, int n_in,
                              void* d_out, int out_size, void* d_ws, size_t ws_size,
                              hipStream_t stream) {
    const float* q      = (const float*)d_in[0];
    const float* knew   = (const float*)d_in[1];
    const float* vnew   = (const float*)d_in[2];
    const float* kcache = (const float*)d_in[3];
    const float* vcache = (const float*)d_in[4];
    const int*   btab   = (const int*)d_in[5];
    const int*   ctxl   = (const int*)d_in[6];
    // d_in[7] = slot_mapping: unused (new token handled in-kernel, caches untouched)

    dim3 grid(B_ * H_);
    dim3 block(NTHREADS);
    pa_decode_kernel<<<grid, block, 0, stream>>>(q, knew, vnew, kcache, vcache,
                                                 btab, ctxl, (float*)d_out);
}